// Conv2D_85822036509358
// MI455X (gfx1250) — compile-verified
//
#include <hip/hip_runtime.h>

typedef float v2f __attribute__((ext_vector_type(2)));
typedef float v8f __attribute__((ext_vector_type(8)));

#define C_IN   16
#define C_OUT  16
#define H_IN   1536
#define W_IN   1536
#define H_OUT  1534
#define W_OUT  1534

#define COLS_PER_WG 128
#define ROWS_PER_WG 4
#define STAGE_COLS  130          // 128 + 2 halo
#define STAGE_ROWS  6            // 4 + 2 halo
#define CI_PAD      18           // 16 ci + 2 pad (bank-conflict-free stride)

// Block = 256 threads = 8 waves. Block computes 4 output rows x 128 columns.
// Wave: 16-pixel column strip, 4 row-tiles -> 4 independent accumulators,
// 144 fp32 WMMAs per wave, A fragments reused across the 4 rows.
// LDS X layout: XL[(row*STAGE_COLS + col)*CI_PAD + ci]  -> B fragment is a
// single contiguous ds_load_b64 (ci, ci+1 at same row/col).
__global__ __launch_bounds__(256) void conv2d_wmma_f32_kernel(
    const float* __restrict__ X,      // [16, 1536, 1536]
    const float* __restrict__ Wt,     // [16, 16, 3, 3]
    const float* __restrict__ bias,   // [16, 1]
    float* __restrict__ out)          // [16, 1534, 1534]
{
    __shared__ float XL[STAGE_ROWS * STAGE_COLS * CI_PAD];
    __shared__ float WL[36 * 64];

    const int t  = threadIdx.x;
    const int h0 = blockIdx.y * ROWS_PER_WG;    // first output row
    const int w0 = blockIdx.x * COLS_PER_WG;    // first output column

    // ---- Stage X: 16 ci x 6 rows x 130 cols, ci-interleaved, edge-clamped ----
    for (int idx = t; idx < C_IN * STAGE_ROWS * STAGE_COLS; idx += 256) {
        int ci  = idx / (STAGE_ROWS * STAGE_COLS);
        int rem = idx - ci * (STAGE_ROWS * STAGE_COLS);
        int r   = rem / STAGE_COLS;
        int c   = rem - r * STAGE_COLS;
        int gr  = h0 + r; if (gr > H_IN - 1) gr = H_IN - 1;
        int gc  = w0 + c; if (gc > W_IN - 1) gc = W_IN - 1;
        XL[(r * STAGE_COLS + c) * CI_PAD + ci] = X[(ci * H_IN + gr) * W_IN + gc];
    }

    // ---- Stage weights in per-lane A-fragment order ----
    // chunk = (kh*3+kw)*4 + g ; lane L<16: (co=L, ci=4g+{0,1}),
    // lane L>=16: (co=L-16, ci=4g+{2,3}).
    for (int idx = t; idx < 36 * 64; idx += 256) {
        int chunk = idx >> 6;
        int r     = idx & 63;
        int lane  = r >> 1;
        int j     = r & 1;
        int kh    = chunk / 12;
        int kw    = (chunk >> 2) % 3;
        int g     = chunk & 3;
        int co    = lane & 15;
        int ci    = 4 * g + ((lane >> 4) << 1) + j;
        WL[idx] = Wt[((co * C_IN + ci) * 3 + kh) * 3 + kw];
    }
    __syncthreads();

    const int lane  = t & 31;
    const int wave  = t >> 5;
    const int laneN = lane & 15;
    const int hi    = lane >> 4;                 // half-wave select
    const int wcol  = wave * 16 + laneN;         // column within staged slab

    v8f acc[ROWS_PER_WG];
    #pragma unroll
    for (int r = 0; r < ROWS_PER_WG; ++r)
        acc[r] = (v8f){0.f, 0.f, 0.f, 0.f, 0.f, 0.f, 0.f, 0.f};

    // per-lane invariant parts of the LDS dword indices
    const int bLane = wcol * CI_PAD + 2 * hi;    // + (r+kh)*STAGE_COLS*CI_PAD + kw*CI_PAD + 4g
    const int aLane = lane * 2;                  // + chunk*64

    #pragma unroll
    for (int kh = 0; kh < 3; ++kh) {
        #pragma unroll
        for (int kw = 0; kw < 3; ++kw) {
            #pragma unroll
            for (int g = 0; g < 4; ++g) {
                const int chunk = (kh * 3 + kw) * 4 + g;
                v2f a = *(const v2f*)&WL[chunk * 64 + aLane];
                #pragma unroll
                for (int r = 0; r < ROWS_PER_WG; ++r) {
                    v2f b = *(const v2f*)&XL[bLane +
                        ((r + kh) * STAGE_COLS + kw) * CI_PAD + 4 * g];
                    acc[r] = __builtin_amdgcn_wmma_f32_16x16x4_f32(
                        /*neg_a=*/false, a, /*neg_b=*/false, b,
                        /*c_mod=*/(short)0, acc[r],
                        /*reuse_a=*/false, /*reuse_b=*/false);
                }
            }
        }
    }

    // sum(bias) — zeros in practice, but reference adds it
    float bsum = 0.f;
    #pragma unroll
    for (int i = 0; i < C_IN; ++i) bsum += bias[i];

    // ---- Store: D vgpr v -> rows co=v (lanes 0-15) / co=v+8 (lanes 16-31) ----
    const int ow = w0 + wcol;
    #pragma unroll
    for (int r = 0; r < ROWS_PER_WG; ++r) {
        const int h = h0 + r;
        if (h < H_OUT && ow < W_OUT) {
            #pragma unroll
            for (int v = 0; v < 8; ++v) {
                int co = v + 8 * hi;
                out[(co * H_OUT + h) * W_OUT + ow] = acc[r][v] + bsum;
            }
        }
    }
}

extern "C" void kernel_launch(void* const* d_in, const int* in_sizes, int n_in,
                              void* d_out, int out_size, void* d_ws, size_t ws_size,
                              hipStream_t stream) {
    const float* X    = (const float*)d_in[0];
    const float* Wt   = (const float*)d_in[1];
    const float* bias = (const float*)d_in[2];
    float* out        = (float*)d_out;

    dim3 block(256, 1, 1);
    dim3 grid((W_OUT + COLS_PER_WG - 1) / COLS_PER_WG,
              (H_OUT + ROWS_PER_WG - 1) / ROWS_PER_WG, 1);
    conv2d_wmma_f32_kernel<<<grid, block, 0, stream>>>(X, Wt, bias, out);
}